// gru_vos_84009560310377
// MI455X (gfx1250) — compile-verified
//
#include <hip/hip_runtime.h>
#include <hip/hip_bf16.h>

constexpr int Himg = 512, Wimg = 512, Bimg = 8;
constexpr int TILE = 32;
constexpr int HALO = 34;     // TILE + 2
constexpr int XSTR = 36;     // LDS row stride (floats) == TDM tile_dim0
constexpr int NTHR = 256;

typedef unsigned int v4u __attribute__((ext_vector_type(4)));
typedef int          v8i __attribute__((ext_vector_type(8)));
typedef int          v4i __attribute__((ext_vector_type(4)));

__device__ __forceinline__ float fast_sigmoid(float v) {
    return 1.0f / (1.0f + __expf(-v));
}
__device__ __forceinline__ float fast_tanh(float v) {
    // tanh(v) = 1 - 2/(e^{2v}+1); saturates correctly for |v| large
    return 1.0f - 2.0f / (__expf(2.0f * v) + 1.0f);
}

__global__ __launch_bounds__(NTHR)
void gru_vos_fused(const float* __restrict__ flow,
                   const float* __restrict__ h_prev,
                   const float* __restrict__ xin,
                   const float* __restrict__ wxz, const float* __restrict__ bxz,
                   const float* __restrict__ whz, const float* __restrict__ bhz,
                   const float* __restrict__ wxr, const float* __restrict__ bxr,
                   const float* __restrict__ whr, const float* __restrict__ bhr,
                   const float* __restrict__ wxh, const float* __restrict__ bxh,
                   const float* __restrict__ whh, const float* __restrict__ bhh,
                   float* __restrict__ out)
{
    __shared__ float sx[HALO * XSTR];   // x tile: rows y0c.., cols x0c.. (TDM-filled)
    __shared__ float sh[HALO * XSTR];   // warped-h tile: rows y0-1.., cols x0-1..
    __shared__ float swt[60];           // 6x9 weights + 6 biases

    const int tid = threadIdx.x;
    const int b  = blockIdx.z;
    const int x0 = blockIdx.x * TILE;
    const int y0 = blockIdx.y * TILE;
    const int x0c = (x0 > 0) ? (x0 - 1) : 0;   // clamped tile origin (halo)
    const int y0c = (y0 > 0) ? (y0 - 1) : 0;

    // ---------- issue async TDM load of the x halo tile into LDS (wave 0) ----------
    if (tid < 32) {
        unsigned long long gaddr =
            (unsigned long long)(const void*)(xin + ((size_t)b * Himg + y0c) * Wimg + x0c);
        unsigned int ldsoff = (unsigned int)(size_t)(void*)&sx[0];

        v4u g0;
        g0[0] = 1u;                                        // count=1, user descriptor
        g0[1] = ldsoff;                                    // lds_addr
        g0[2] = (unsigned int)(gaddr & 0xFFFFFFFFu);       // global_addr lo
        g0[3] = (unsigned int)((gaddr >> 32) & 0x1FFFFFFu) // global_addr hi (57b)
              | (2u << 30);                                // type = 2 (image)

        unsigned int td0 = (unsigned int)(Wimg - x0c);     // remaining cols (OOB -> 0)
        unsigned int td1 = (unsigned int)(Himg - y0c);     // remaining rows (OOB -> 0)
        unsigned long long st1 = (unsigned long long)Wimg * Himg;

        v8i g1;
        g1[0] = (int)(2u << 16);                                        // data_size = 4B
        g1[1] = (int)((td0 & 0xFFFFu) << 16);                           // tensor_dim0[15:0]
        g1[2] = (int)(((td0 >> 16) & 0xFFFFu) | ((td1 & 0xFFFFu) << 16));
        g1[3] = (int)(((td1 >> 16) & 0xFFFFu) | ((unsigned)XSTR << 16)); // tile_dim0 = 36
        g1[4] = (int)((unsigned)HALO);                                  // tile_dim1=34, tile_dim2=0
        g1[5] = (int)Wimg;                                              // dim0 stride lo
        g1[6] = (int)((st1 & 0xFFFFull) << 16);                         // dim1 stride[15:0]
        g1[7] = (int)(st1 >> 16);                                       // dim1 stride hi
        v4i zz4 = {0, 0, 0, 0};
        v8i zz8 = {0, 0, 0, 0, 0, 0, 0, 0};
        __builtin_amdgcn_tensor_load_to_lds(g0, g1, zz4, zz4, zz8, 0);
    }

    // ---------- stage weights/biases into LDS ----------
    if (tid < 54) {
        const float* wp[6] = {wxz, whz, wxr, whr, wxh, whh};
        swt[tid] = wp[tid / 9][tid % 9];
    } else if (tid < 60) {
        const float* bp[6] = {bxz, bhz, bxr, bhr, bxh, bhh};
        swt[tid] = bp[tid - 54][0];
    }

    // ---------- compute warped h for the 34x34 halo tile (overlaps TDM) ----------
    {
        const float* hp  = h_prev + (size_t)b * Himg * Wimg;
        const float* fyp = flow + ((size_t)b * 2 + 0) * (size_t)Himg * Wimg;
        const float* fxp = flow + ((size_t)b * 2 + 1) * (size_t)Himg * Wimg;
        for (int s = tid; s < HALO * HALO; s += NTHR) {
            int r = s / HALO, c = s % HALO;
            int gy = y0 - 1 + r, gx = x0 - 1 + c;
            float hv = 0.0f;
            if (gy >= 0 && gy < Himg && gx >= 0 && gx < Wimg) {
                size_t p = (size_t)gy * Wimg + gx;
                float yy = (float)gy + fyp[p];
                float xx = (float)gx + fxp[p];
                float yf = floorf(yy), xf = floorf(xx);
                float wy = yy - yf,   wx = xx - xf;
                int yi = (int)yf,     xi = (int)xf;
                auto corner = [&](int yq, int xq) -> float {
                    bool v = (yq >= 0) & (yq < Himg) & (xq >= 0) & (xq < Wimg);
                    int yc = min(max(yq, 0), Himg - 1);
                    int xc = min(max(xq, 0), Wimg - 1);
                    float val = hp[(size_t)yc * Wimg + xc];
                    return v ? val : 0.0f;
                };
                float v00 = corner(yi,     xi);
                float v01 = corner(yi,     xi + 1);
                float v10 = corner(yi + 1, xi);
                float v11 = corner(yi + 1, xi + 1);
                hv = v00 * (1.0f - wy) * (1.0f - wx) + v01 * (1.0f - wy) * wx
                   + v10 * wy * (1.0f - wx)          + v11 * wy * wx;
            }
            sh[r * XSTR + c] = hv;
        }
    }

    __builtin_amdgcn_s_wait_tensorcnt(0);   // TDM tile resident in LDS (wave 0)
    __syncthreads();                        // publish sx/sh/swt to all waves

    // ---------- per-thread: 4 vertically adjacent pixels at one column ----------
    const int lx  = tid & 31;
    const int ty  = tid >> 5;
    const int gx  = x0 + lx;
    const int gyb = y0 + ty * 4;

    float xp[6][3], hq[6][3];

    // warped-h tile always covers rows y0-1..y0+32 / cols x0-1..x0+32 with
    // explicit zeros at OOB slots, so no checks needed.
    #pragma unroll
    for (int r = 0; r < 6; ++r)
        #pragma unroll
        for (int c = 0; c < 3; ++c)
            hq[r][c] = sh[(ty * 4 + r) * XSTR + (lx + c)];

    // x tile: strictly interior blocks (uniform branch) need no per-element
    // checks — every neighborhood coordinate is inside the tensor, hence
    // TDM-written. Edge blocks take the fully-checked path (never reads an
    // LDS slot whose coordinate is out of tensor).
    const bool interior = (x0 > 0) & (y0 > 0) & (x0 + TILE < Wimg) & (y0 + TILE < Himg);
    if (interior) {
        #pragma unroll
        for (int r = 0; r < 6; ++r)
            #pragma unroll
            for (int c = 0; c < 3; ++c)
                xp[r][c] = sx[(ty * 4 + r) * XSTR + (lx + c)];
    } else {
        #pragma unroll
        for (int r = 0; r < 6; ++r) {
            int gy = gyb - 1 + r;
            #pragma unroll
            for (int c = 0; c < 3; ++c) {
                int gxx = gx - 1 + c;
                float xv = 0.0f;
                if (gy >= 0 && gy < Himg && gxx >= 0 && gxx < Wimg)
                    xv = sx[(gy - y0c) * XSTR + (gxx - x0c)];
                xp[r][c] = xv;
            }
        }
    }

    float Wl[54];
    #pragma unroll
    for (int k = 0; k < 54; ++k) Wl[k] = swt[k];
    const float Bxz = swt[54], Bhz = swt[55], Bxr = swt[56];
    const float Bhr = swt[57], Bxh = swt[58], Bhh = swt[59];

    float* op = out + (size_t)b * Himg * Wimg;
    #pragma unroll
    for (int i = 0; i < 4; ++i) {
        float axz = 0.f, ahz = 0.f, axr = 0.f, ahr = 0.f, axh = 0.f, ahh = 0.f;
        #pragma unroll
        for (int dy = 0; dy < 3; ++dy) {
            #pragma unroll
            for (int dx = 0; dx < 3; ++dx) {
                const int k = dy * 3 + dx;
                const float xn = xp[i + dy][dx];
                const float hn = hq[i + dy][dx];
                axz = fmaf(Wl[k],      xn, axz);
                ahz = fmaf(Wl[9 + k],  hn, ahz);
                axr = fmaf(Wl[18 + k], xn, axr);
                ahr = fmaf(Wl[27 + k], hn, ahr);
                axh = fmaf(Wl[36 + k], xn, axh);
                ahh = fmaf(Wl[45 + k], hn, ahh);
            }
        }
        const float z  = fast_sigmoid(axz + Bxz + ahz + Bhz);
        const float r  = fast_sigmoid(axr + Bxr + ahr + Bhr);
        const float hc = fast_tanh(axh + Bxh + r * (ahh + Bhh));
        const float hcen = hq[i + 1][1];
        op[(size_t)(gyb + i) * Wimg + gx] = (1.0f - z) * hcen + z * hc;
    }
}

extern "C" void kernel_launch(void* const* d_in, const int* in_sizes, int n_in,
                              void* d_out, int out_size, void* d_ws, size_t ws_size,
                              hipStream_t stream) {
    // setup_inputs order:
    // 0 image (unused), 1 image_ref (unused), 2 flow, 3 h_prev, 4 unary_input,
    // 5 wxz, 6 bxz, 7 whz, 8 bhz, 9 wxr, 10 bxr, 11 whr, 12 bhr,
    // 13 wxh, 14 bxh, 15 whh, 16 bhh
    const float* flow   = (const float*)d_in[2];
    const float* h_prev = (const float*)d_in[3];
    const float* xin    = (const float*)d_in[4];
    dim3 grid(Wimg / TILE, Himg / TILE, Bimg);
    gru_vos_fused<<<grid, NTHR, 0, stream>>>(
        flow, h_prev, xin,
        (const float*)d_in[5],  (const float*)d_in[6],
        (const float*)d_in[7],  (const float*)d_in[8],
        (const float*)d_in[9],  (const float*)d_in[10],
        (const float*)d_in[11], (const float*)d_in[12],
        (const float*)d_in[13], (const float*)d_in[14],
        (const float*)d_in[15], (const float*)d_in[16],
        (float*)d_out);
}